// GC_withres_61272003444924
// MI455X (gfx1250) — compile-verified
//
#include <hip/hip_runtime.h>

#define D_FEAT 256
#define SMOOTH 0.5f

typedef __attribute__((ext_vector_type(2))) float v2f;
typedef __attribute__((ext_vector_type(8))) float v8f;

// ---------------- init: deg = 1.0 (identity in A_gcn), Zacc = 0 ----------------
__global__ __launch_bounds__(256) void init_kernel(float* __restrict__ deg,
                                                   float* __restrict__ Zacc,
                                                   int N, long long total) {
    long long i = (long long)blockIdx.x * blockDim.x + threadIdx.x;
    if (i < total) Zacc[i] = 0.0f;
    if (i < N)     deg[i]  = 1.0f;
}

// ---------------- deg[col] += w ----------------
__global__ __launch_bounds__(256) void deg_kernel(const float* __restrict__ ew,
                                                  const int* __restrict__ ec,
                                                  float* __restrict__ deg, int E) {
    int i = blockIdx.x * blockDim.x + threadIdx.x;
    if (i < E) atomicAdd(&deg[ec[i]], ew[i]);
}

// ---------------- Dm12 = deg^-1/2 (in place) ----------------
__global__ __launch_bounds__(256) void rsqrt_kernel(float* __restrict__ deg, int N) {
    int i = blockIdx.x * blockDim.x + threadIdx.x;
    if (i < N) deg[i] = rsqrtf(deg[i]);
}

// ---------------- support = x @ W + b via V_WMMA_F32_16X16X4_F32 ----------------
// One wave computes a 16x16 tile of support. K swept in steps of 4.
// A (16x4 f32): lanes 0-15 -> rows 0-15 holding {K, K+1}; lanes 16-31 -> {K+2, K+3}
// B (4x16 f32): lanes 0-15 -> cols, VGPR0=K row, VGPR1=K+1; lanes 16-31 -> K+2/K+3
// C/D (16x16 f32): VGPR j: lanes 0-15 -> M=j, lanes 16-31 -> M=j+8
__global__ __launch_bounds__(256) void gemm_wmma_kernel(const float* __restrict__ x,
                                                        const float* __restrict__ W,
                                                        const float* __restrict__ bias,
                                                        float* __restrict__ support,
                                                        int Nrows) {
    const int wave = (blockIdx.x * blockDim.x + threadIdx.x) >> 5;
    const int lane = threadIdx.x & 31;

    const int tilesN = D_FEAT / 16;              // 16 column tiles
    const int tileM  = (wave / tilesN) * 16;
    const int tileN  = (wave % tilesN) * 16;
    if (tileM >= Nrows) return;

    const int half16 = lane & 15;                // row for A / col for B
    const int khalf  = (lane >> 4) * 2;          // 0 for lanes 0-15, 2 for 16-31

    const float* __restrict__ xrow = x + (long long)(tileM + half16) * D_FEAT;
    const float* __restrict__ wcol = W + (tileN + half16);

    v8f c = {};
#pragma unroll
    for (int k = 0; k < D_FEAT; k += 4) {
        v2f a, bmat;
        a.x    = xrow[k + khalf];
        a.y    = xrow[k + khalf + 1];
        bmat.x = wcol[(long long)(k + khalf)     * D_FEAT];
        bmat.y = wcol[(long long)(k + khalf + 1) * D_FEAT];
        // (neg_a, A, neg_b, B, c_mod, C, reuse_a, reuse_b)
        c = __builtin_amdgcn_wmma_f32_16x16x4_f32(false, a, false, bmat,
                                                  (short)0, c, false, false);
    }

    const int colg  = tileN + half16;
    const float bv  = bias[colg];
    const int mhalf = (lane >> 4) * 8;
#pragma unroll
    for (int j = 0; j < 8; ++j) {
        const int m = tileM + mhalf + j;
        support[(long long)m * D_FEAT + colg] = c[j] + bv;
    }
}

// ---------------- SpMM scatter: Zacc[row] += (w * Dm12[col]) * support[col] ----------------
// One wave32 per edge; lane-strided float4 loads (coalesced 128B), f32 global atomics.
__global__ __launch_bounds__(256) void scatter_kernel(const float* __restrict__ support,
                                                      const float* __restrict__ Dm12,
                                                      const float* __restrict__ ew,
                                                      const int* __restrict__ er,
                                                      const int* __restrict__ ec,
                                                      float* __restrict__ Zacc, int E) {
    const int edge = (blockIdx.x * blockDim.x + threadIdx.x) >> 5;
    const int lane = threadIdx.x & 31;
    if (edge >= E) return;

    const int   row   = er[edge];
    const int   col   = ec[edge];
    const float coeff = ew[edge] * Dm12[col];

    const float4* __restrict__ src = (const float4*)(support + (long long)col * D_FEAT);
    float* __restrict__ dst = Zacc + (long long)row * D_FEAT;

#pragma unroll
    for (int h = 0; h < 2; ++h) {
        const int q = lane + 32 * h;             // float4 index: 64 per row
        const float4 v = src[q];
        const int d = q * 4;
        atomicAdd(dst + d + 0, coeff * v.x);
        atomicAdd(dst + d + 1, coeff * v.y);
        atomicAdd(dst + d + 2, coeff * v.z);
        atomicAdd(dst + d + 3, coeff * v.w);
    }
}

// ---------------- finalize: out = (Dm12*(Z + s*Dm12)*SMOOTH + s)/(1+SMOOTH), in place ----
__global__ __launch_bounds__(256) void finalize_kernel(float* __restrict__ support_out,
                                                       const float* __restrict__ Zacc,
                                                       const float* __restrict__ Dm12,
                                                       long long ND) {
    long long i = (long long)blockIdx.x * blockDim.x + threadIdx.x;
    if (i >= ND) return;
    const int   r  = (int)(i >> 8);              // D_FEAT == 256
    const float dm = Dm12[r];
    const float s  = support_out[i];
    const float z  = dm * (Zacc[i] + s * dm);
    support_out[i] = (z * SMOOTH + s) * (1.0f / (1.0f + SMOOTH));
}

extern "C" void kernel_launch(void* const* d_in, const int* in_sizes, int n_in,
                              void* d_out, int out_size, void* d_ws, size_t ws_size,
                              hipStream_t stream) {
    const float* x  = (const float*)d_in[0];
    const float* W  = (const float*)d_in[1];
    const float* b  = (const float*)d_in[2];
    const float* ew = (const float*)d_in[3];
    const int*   er = (const int*)d_in[4];
    const int*   ec = (const int*)d_in[5];
    float* out = (float*)d_out;                  // holds `support`, then final result

    const int N = in_sizes[0] / D_FEAT;          // 100000
    const int E = in_sizes[3];                   // 3200000
    const long long ND = (long long)N * D_FEAT;

    float* deg  = (float*)d_ws;                  // N floats (becomes Dm12)
    float* Zacc = deg + N;                       // N*D floats (offset 400000B, 16B aligned)

    // 1) init deg=1, Zacc=0
    {
        long long total = ND;
        int blocks = (int)((total + 255) / 256);
        init_kernel<<<blocks, 256, 0, stream>>>(deg, Zacc, N, total);
    }
    // 2) degree accumulation
    deg_kernel<<<(E + 255) / 256, 256, 0, stream>>>(ew, ec, deg, E);
    // 3) Dm12 = deg^-1/2
    rsqrt_kernel<<<(N + 255) / 256, 256, 0, stream>>>(deg, N);
    // 4) support = x@W + b  (WMMA f32), written into d_out
    {
        int tiles = ((N + 15) / 16) * (D_FEAT / 16);   // waves
        int blocks = (tiles + 7) / 8;                  // 8 waves per 256-thread block
        gemm_wmma_kernel<<<blocks, 256, 0, stream>>>(x, W, b, out, N);
    }
    // 5) edge scatter into Zacc
    {
        int blocks = (E + 7) / 8;                      // one wave per edge
        scatter_kernel<<<blocks, 256, 0, stream>>>(out, deg, ew, er, ec, Zacc, E);
    }
    // 6) finalize in place in d_out
    {
        int blocks = (int)((ND + 255) / 256);
        finalize_kernel<<<blocks, 256, 0, stream>>>(out, Zacc, deg, ND);
    }
}